// MHGAT_20091857010731
// MI455X (gfx1250) — compile-verified
//
#include <hip/hip_runtime.h>
#include <stdint.h>

#define TPB 256

#ifndef __has_builtin
#define __has_builtin(x) 0
#endif
#if __has_builtin(__builtin_amdgcn_global_load_async_to_lds_b32) && \
    __has_builtin(__builtin_amdgcn_s_wait_asynccnt)
#define USE_ASYNC_LDS 1
#else
#define USE_ASYNC_LDS 0
#endif

// ---- problem constants (from reference) ----
constexpr int NB   = 32;     // batch
constexpr int NN   = 1024;   // nodes
constexpr int NE   = 16384;  // edges
constexpr int NH   = 4;      // heads
constexpr int NC   = 64;     // channels/head
constexpr int NHC  = 256;    // H*C
constexpr int NFC0 = 128;
constexpr float SLOPE = 0.2f;

// ---- workspace layout (float elements, per batch) ----
constexpr size_t OFF_H     = 0;                           // N*HC transformed features (reused for FC0 activations)
constexpr size_t OFF_Y     = OFF_H   + (size_t)NN * NHC;  // layer output / residual carrier
constexpr size_t OFF_AGG   = OFF_Y   + (size_t)NN * NHC;  // edge-aggregation accumulator
constexpr size_t OFF_AS    = OFF_AGG + (size_t)NN * NHC;  // a_src (N,H)
constexpr size_t OFF_AD    = OFF_AS  + (size_t)NN * NH;   // a_dst (N,H)
constexpr size_t OFF_M     = OFF_AD  + (size_t)NN * NH;   // segment max, monotone u32 key
constexpr size_t OFF_S     = OFF_M   + (size_t)NN * NH;   // segment sum of exp
constexpr size_t OFF_ALPHA = OFF_S   + (size_t)NN * NH;   // per-edge logits / exp (E,H)
constexpr size_t WS_B      = OFF_ALPHA + (size_t)NE * NH; // per-batch stride (floats)

constexpr int LDS_PAD = 4;                 // rows stride K+4 -> lanes hit distinct banks
constexpr int LDS_ELEMS = 16 * (256 + LDS_PAD);

typedef __attribute__((ext_vector_type(2))) float v2f;
typedef __attribute__((ext_vector_type(8))) float v8f;

#if USE_ASYNC_LDS
typedef __attribute__((address_space(1))) int as1_int;
typedef __attribute__((address_space(3))) int as3_int;
#endif

// monotone float -> u32 key (order-preserving over all finite floats)
__device__ __forceinline__ unsigned fkey(float v) {
    unsigned u = __float_as_uint(v);
    return (u & 0x80000000u) ? ~u : (u | 0x80000000u);
}
__device__ __forceinline__ float fdec(unsigned k) {
    return (k & 0x80000000u) ? __uint_as_float(k ^ 0x80000000u) : __uint_as_float(~k);
}

// ---------------- zero accumulators (agg, m, s) ----------------
__global__ void mhgat_zero(float* __restrict__ ws) {
    float* base = ws + (size_t)blockIdx.y * WS_B;
    int gid = blockIdx.x * TPB + threadIdx.x;      // covers N*HC
    base[OFF_AGG + gid] = 0.0f;
    if (gid < NN * NH) {
        ((unsigned*)(base + OFF_M))[gid] = 0u;     // key-space -inf
        base[OFF_S + gid] = 0.0f;
    }
}

// ---------------- WMMA f32 16x16x4 GEMM: C = A*B (+bias, relu) -------------
// block = 8 wave32 = one 16x128 strip of C (one m-tile, 8 n-tiles).
// A tile (16 x K) is staged once per block in LDS via gfx1250 async-to-LDS
// (ASYNCcnt-tracked), shared by all 8 waves.
__global__ void mhgat_gemm_wmma(const float* __restrict__ A, int lda, size_t aStride,
                                const float* __restrict__ Bm, int ldb,
                                float* __restrict__ C, int ldc, size_t cStride,
                                int K, int kLog2, int nColBlks,
                                const float* __restrict__ bias, int relu) {
    __shared__ float atile[LDS_ELEMS];
    const int  wave = threadIdx.x >> 5;
    const int  lane = threadIdx.x & 31;
    const int  m0   = (blockIdx.x / nColBlks) << 4;
    const int  n0   = (blockIdx.x % nColBlks) * 128 + (wave << 4);
    const int  l16  = lane & 15;
    const int  half = lane >> 4;
    const int  stride = K + LDS_PAD;
    const float* Ab = A + (size_t)blockIdx.y * aStride;
    float*       Cb = C + (size_t)blockIdx.y * cStride;

    // ---- stage A tile (16 rows x K) into LDS, cooperatively ----
    const int total = K << 4;                       // 16*K elements
    for (int idx = threadIdx.x; idx < total; idx += TPB) {
        const int r = idx >> kLog2;                 // K is a power of two (4 or 256)
        const int k = idx & (K - 1);
#if USE_ASYNC_LDS
        __builtin_amdgcn_global_load_async_to_lds_b32(
            (as1_int*)(uintptr_t)(Ab + (size_t)(m0 + r) * lda + k),
            (as3_int*)(uint32_t)(uintptr_t)(&atile[r * stride + k]),
            0, 0);
#else
        atile[r * stride + k] = Ab[(size_t)(m0 + r) * lda + k];
#endif
    }
#if USE_ASYNC_LDS
    __builtin_amdgcn_s_wait_asynccnt(0);
#endif
    __syncthreads();

    const int col = n0 + l16;                       // B/C column for this lane
    const float* bp0 = Bm + (size_t)(half << 1) * ldb + col;   // B row kb
    const float* bp1 = bp0 + ldb;                               // B row kb+1
    __builtin_prefetch(bp0, 0, 1);                  // -> global_prefetch_b8
    const int rowBase = l16 * stride;

    v8f acc = {0.f, 0.f, 0.f, 0.f, 0.f, 0.f, 0.f, 0.f};
    int kb = half << 1;                             // lanes 0-15: k0,k0+1 ; 16-31: k0+2,k0+3
    for (int k0 = 0; k0 < K; k0 += 4) {
        v2f a, b;
        a.x = atile[rowBase + kb];                  // ds_load (shared A tile)
        a.y = atile[rowBase + kb + 1];
        b.x = *bp0;
        b.y = *bp1;
        acc = __builtin_amdgcn_wmma_f32_16x16x4_f32(false, a, false, b,
                                                    (short)0, acc, false, false);
        kb  += 4;
        bp0 += (size_t)ldb << 2;
        bp1 += (size_t)ldb << 2;
    }
    const float bb = bias ? bias[col] : 0.0f;
#pragma unroll
    for (int r = 0; r < 8; ++r) {
        float v = acc[r] + bb;
        if (relu) v = fmaxf(v, 0.0f);
        Cb[(size_t)(m0 + r + half * 8) * ldc + col] = v;   // C: M = r + 8*half
    }
}

// ---------------- attention coefficients: aS/aD = <h[n,h,:], att[h,:]> ------
__global__ void mhgat_attn(float* __restrict__ ws,
                           const float* __restrict__ att_src,
                           const float* __restrict__ att_dst) {
    float* base = ws + (size_t)blockIdx.y * WS_B;
    int gid = blockIdx.x * TPB + threadIdx.x;      // covers N*H
    int n = gid >> 2, hh = gid & 3;
    const float* hr = base + OFF_H + (size_t)n * NHC + hh * NC;
    const float* as = att_src + hh * NC;
    const float* ad = att_dst + hh * NC;
    float s0 = 0.f, s1 = 0.f;
#pragma unroll 4
    for (int c = 0; c < NC; ++c) {
        float v = hr[c];
        s0 += v * as[c];
        s1 += v * ad[c];
    }
    base[OFF_AS + gid] = s0;
    base[OFF_AD + gid] = s1;
}

// ---------------- edge pass 1: logits + segment max (atomic, keyed) --------
__global__ void mhgat_edge_logit_max(float* __restrict__ ws,
                                     const int* __restrict__ ei) {
    float* base = ws + (size_t)blockIdx.y * WS_B;
    int e   = blockIdx.x * TPB + threadIdx.x;      // covers E
    int src = ei[e], dst = ei[NE + e];
    unsigned* mbuf = (unsigned*)(base + OFF_M);
#pragma unroll
    for (int hh = 0; hh < NH; ++hh) {
        float l = base[OFF_AS + src * NH + hh] + base[OFF_AD + dst * NH + hh];
        l = (l > 0.0f) ? l : SLOPE * l;            // leaky relu
        base[OFF_ALPHA + (size_t)e * NH + hh] = l;
        atomicMax(&mbuf[dst * NH + hh], fkey(l));
    }
}

// ---------------- edge pass 2: exp(logit - max) + segment sum --------------
__global__ void mhgat_edge_exp_sum(float* __restrict__ ws,
                                   const int* __restrict__ ei) {
    float* base = ws + (size_t)blockIdx.y * WS_B;
    int e   = blockIdx.x * TPB + threadIdx.x;
    int dst = ei[NE + e];
    const unsigned* mbuf = (const unsigned*)(base + OFF_M);
#pragma unroll
    for (int hh = 0; hh < NH; ++hh) {
        float m  = fdec(mbuf[dst * NH + hh]);
        float ex = __expf(base[OFF_ALPHA + (size_t)e * NH + hh] - m);
        base[OFF_ALPHA + (size_t)e * NH + hh] = ex;
        unsafeAtomicAdd(&base[OFF_S + dst * NH + hh], ex);
    }
}

// ---------------- edge pass 3: weighted scatter of messages ----------------
__global__ void mhgat_edge_aggregate(float* __restrict__ ws,
                                     const int* __restrict__ ei) {
    float* base = ws + (size_t)blockIdx.y * WS_B;
    int gid = blockIdx.x * TPB + threadIdx.x;      // covers E*NC
    int e = gid >> 6, c = gid & (NC - 1);
    int src = ei[e], dst = ei[NE + e];
#pragma unroll
    for (int hh = 0; hh < NH; ++hh) {
        float w = base[OFF_ALPHA + (size_t)e * NH + hh] /
                  (base[OFF_S + dst * NH + hh] + 1e-16f);
        float v = base[OFF_H + (size_t)src * NHC + hh * NC + c] * w;
        unsafeAtomicAdd(&base[OFF_AGG + (size_t)dst * NHC + hh * NC + c], v);
    }
}

// ---------------- bias (+ optional residual): y = [y +] agg + b ------------
__global__ void mhgat_bias_res(float* __restrict__ ws,
                               const float* __restrict__ bias, int residual) {
    float* base = ws + (size_t)blockIdx.y * WS_B;
    int gid = blockIdx.x * TPB + threadIdx.x;      // covers N*HC
    float v = base[OFF_AGG + gid] + bias[gid & (NHC - 1)];
    if (residual) v += base[OFF_Y + gid];
    base[OFF_Y + gid] = v;
}

// ---------------- FC1: out[n] = <z[n,:], w> + b ----------------------------
__global__ void mhgat_fc1(const float* __restrict__ ws,
                          const float* __restrict__ w,
                          const float* __restrict__ b,
                          float* __restrict__ out) {
    const float* base = ws + (size_t)blockIdx.y * WS_B;
    int n = blockIdx.x * TPB + threadIdx.x;        // covers N
    const float* z = base + OFF_H + (size_t)n * NFC0;   // FC0 activations alias OFF_H
    float acc = b[0];
#pragma unroll 4
    for (int j = 0; j < NFC0; ++j) acc += z[j] * w[j];
    out[(size_t)blockIdx.y * NN + n] = acc;
}

extern "C" void kernel_launch(void* const* d_in, const int* in_sizes, int n_in,
                              void* d_out, int out_size, void* d_ws, size_t ws_size,
                              hipStream_t stream) {
    const float* x        = (const float*)d_in[0];
    // d_in[1] = e, ignored (edge_dim=None in reference)
    const int*   ei0      = (const int*)d_in[2];
    const int*   ei1      = (const int*)d_in[3];
    const float* W0       = (const float*)d_in[4];
    const float* att_src0 = (const float*)d_in[5];
    const float* att_dst0 = (const float*)d_in[6];
    const float* b0       = (const float*)d_in[7];
    const float* W1       = (const float*)d_in[8];
    const float* att_src1 = (const float*)d_in[9];
    const float* att_dst1 = (const float*)d_in[10];
    const float* b1       = (const float*)d_in[11];
    const float* fc0_w    = (const float*)d_in[12];
    const float* fc0_b    = (const float*)d_in[13];
    const float* fc1_w    = (const float*)d_in[14];
    const float* fc1_b    = (const float*)d_in[15];
    float* out = (float*)d_out;
    float* ws  = (float*)d_ws;

    const dim3 blk(TPB);
    const dim3 gZero(NN * NHC / TPB, NB);          // 1024 x 32
    const dim3 gAttn(NN * NH / TPB, NB);           // 16 x 32
    const dim3 gEdge(NE / TPB, NB);                // 64 x 32
    const dim3 gAggr(NE * NC / TPB, NB);           // 4096 x 32
    const dim3 gBias(NN * NHC / TPB, NB);          // 1024 x 32
    const dim3 gGemmL((NN / 16) * (NHC / 128), NB);  // 128 x 32 : 16x128 strip per block
    const dim3 gGemmF((NN / 16) * (NFC0 / 128), NB); // 64 x 32
    const dim3 gFc1(NN / TPB, NB);                 // 4 x 32

    // ---- GAT layer 0: h = x @ W0 (K=4, one WMMA per tile) ----
    mhgat_zero<<<gZero, blk, 0, stream>>>(ws);
    mhgat_gemm_wmma<<<gGemmL, blk, 0, stream>>>(x, 4, (size_t)NN * 4,
                                                W0, NHC,
                                                ws + OFF_H, NHC, WS_B,
                                                4, 2, NHC / 128, nullptr, 0);
    mhgat_attn<<<gAttn, blk, 0, stream>>>(ws, att_src0, att_dst0);
    mhgat_edge_logit_max<<<gEdge, blk, 0, stream>>>(ws, ei0);
    mhgat_edge_exp_sum<<<gEdge, blk, 0, stream>>>(ws, ei0);
    mhgat_edge_aggregate<<<gAggr, blk, 0, stream>>>(ws, ei0);
    mhgat_bias_res<<<gBias, blk, 0, stream>>>(ws, b0, /*residual=*/0);

    // ---- GAT layer 1 (+ residual): h = y @ W1 (K=256) ----
    mhgat_zero<<<gZero, blk, 0, stream>>>(ws);
    mhgat_gemm_wmma<<<gGemmL, blk, 0, stream>>>(ws + OFF_Y, NHC, WS_B,
                                                W1, NHC,
                                                ws + OFF_H, NHC, WS_B,
                                                NHC, 8, NHC / 128, nullptr, 0);
    mhgat_attn<<<gAttn, blk, 0, stream>>>(ws, att_src1, att_dst1);
    mhgat_edge_logit_max<<<gEdge, blk, 0, stream>>>(ws, ei1);
    mhgat_edge_exp_sum<<<gEdge, blk, 0, stream>>>(ws, ei1);
    mhgat_edge_aggregate<<<gAggr, blk, 0, stream>>>(ws, ei1);
    mhgat_bias_res<<<gBias, blk, 0, stream>>>(ws, b1, /*residual=*/1);

    // ---- FC head: z = relu(y @ fc0_w + fc0_b) ; out = z @ fc1_w + fc1_b ----
    mhgat_gemm_wmma<<<gGemmF, blk, 0, stream>>>(ws + OFF_Y, NHC, WS_B,
                                                fc0_w, NFC0,
                                                ws + OFF_H, NFC0, WS_B,   // z aliases OFF_H
                                                NHC, 8, NFC0 / 128, fc0_b, /*relu=*/1);
    mhgat_fc1<<<gFc1, blk, 0, stream>>>(ws, fc1_w, fc1_b, out);
}